// Transformer_37709812859296
// MI455X (gfx1250) — compile-verified
//
#include <hip/hip_runtime.h>
#include <hip/hip_bf16.h>

typedef float v2f __attribute__((ext_vector_type(2)));
typedef float v8f __attribute__((ext_vector_type(8)));
typedef float v4f __attribute__((ext_vector_type(4)));
typedef int   v4i __attribute__((ext_vector_type(4)));

#define GPAD 4
#define AS1 __attribute__((address_space(1)))
#define AS3 __attribute__((address_space(3)))

#if defined(__has_builtin)
#if __has_builtin(__builtin_amdgcn_global_load_async_to_lds_b128)
#define HAVE_ASYNC_LDS 1
#endif
#endif

__device__ __forceinline__ void wait_async_lds()
{
#if defined(HAVE_ASYNC_LDS)
#if __has_builtin(__builtin_amdgcn_s_wait_asynccnt)
    __builtin_amdgcn_s_wait_asynccnt(0);
#else
    asm volatile("s_wait_asynccnt 0x0" ::: "memory");
#endif
#endif
}

// Copy one row segment global -> LDS (per-lane), async if available.
__device__ __forceinline__ void copy_row_to_lds(const float* __restrict__ gsrc,
                                                float* __restrict__ ldst,
                                                int Kdim, int lane)
{
#if defined(HAVE_ASYNC_LDS)
    for (int c4 = lane; c4 < (Kdim >> 2); c4 += 32) {
        __builtin_amdgcn_global_load_async_to_lds_b128(
            (AS1 v4i*)(gsrc + c4 * 4), (AS3 v4i*)(ldst + c4 * 4), 0, 0);
    }
#else
    for (int c4 = lane; c4 < (Kdim >> 2); c4 += 32) {
        *(v4f*)(ldst + c4 * 4) = *(const v4f*)(gsrc + c4 * 4);
    }
#endif
}

// ---------------------------------------------------------------------------
// fp32 WMMA GEMM: C[M,N] = (A + A2?)[M,K] @ W[N,K]^T + bias, opt ReLU.
// One block (8 waves) owns one 16-row M tile. A (and A2) tiles async-copied
// into LDS once; each wave sweeps N tiles two-at-a-time (tn, tn+8) with two
// independent v_wmma_f32_16x16x4_f32 accumulator chains sharing the A frag.
// HASA2 is a template parameter so the hot K-loop has no runtime branches.
// ---------------------------------------------------------------------------
template <bool HASA2>
__global__ __launch_bounds__(256) void gemm_wmma_f32(
    const float* __restrict__ A, const float* __restrict__ A2,
    const float* __restrict__ W, const float* __restrict__ bias,
    float* __restrict__ C, int Mrows, int Kdim, int Ndim, int relu)
{
    extern __shared__ float lds[];
    const int lane = threadIdx.x & 31;
    const int wave = threadIdx.x >> 5;
    const int tm   = blockIdx.x;
    const int ldst = Kdim + GPAD;
    float* ldsA  = lds;
    float* ldsA2 = lds + 16 * ldst;
    const long abase = (long)tm * 16 * Kdim;

    // Stage A tile (and A2 tile) into LDS: wave w copies rows w and w+8.
    for (int r = wave; r < 16; r += 8) {
        copy_row_to_lds(A + abase + (long)r * Kdim, ldsA + r * ldst, Kdim, lane);
        if (HASA2)
            copy_row_to_lds(A2 + abase + (long)r * Kdim, ldsA2 + r * ldst, Kdim, lane);
    }
    wait_async_lds();
    __syncthreads();

    const int tilesN = Ndim >> 4;
    const int khalf  = (lane >> 4) << 1;          // lanes 16-31 hold K+2,K+3
    const float* alsA  = ldsA  + (lane & 15) * ldst + khalf;
    const float* alsA2 = ldsA2 + (lane & 15) * ldst + khalf;

    const int coll  = lane & 15;
    const int rbase = tm * 16 + ((lane >> 4) << 3); // lanes>=16 -> rows +8

    auto epilogue = [&](v8f c, int tn_) {
        const int coln = tn_ * 16 + coll;
        const float bv = bias ? bias[coln] : 0.f;
#pragma unroll
        for (int r = 0; r < 8; r++) {
            float v = c[r] + bv;
            if (relu) v = fmaxf(v, 0.f);
            C[(long)(rbase + r) * Ndim + coln] = v;
        }
    };

    int tn = wave;
    for (; tn + 8 < tilesN; tn += 16) {          // paired N tiles: tn, tn+8
        const float* w0 = W + (long)(tn * 16 + coll) * Kdim + khalf;
        const float* w1 = w0 + (long)128 * Kdim; // +8 tiles = +128 rows of W
        v8f c0 = {0.f, 0.f, 0.f, 0.f, 0.f, 0.f, 0.f, 0.f};
        v8f c1 = {0.f, 0.f, 0.f, 0.f, 0.f, 0.f, 0.f, 0.f};
#pragma unroll 8
        for (int k = 0; k < Kdim; k += 4) {
            v2f a = *(const v2f*)(alsA + k);
            if (HASA2) a += *(const v2f*)(alsA2 + k);
            v2f b0 = *(const v2f*)(w0 + k);
            v2f b1 = *(const v2f*)(w1 + k);
            c0 = __builtin_amdgcn_wmma_f32_16x16x4_f32(
                     false, a, false, b0, (short)0, c0, false, false);
            c1 = __builtin_amdgcn_wmma_f32_16x16x4_f32(
                     false, a, false, b1, (short)0, c1, false, false);
        }
        epilogue(c0, tn);
        epilogue(c1, tn + 8);
    }
    if (tn < tilesN) {                            // remainder single tile
        const float* w0 = W + (long)(tn * 16 + coll) * Kdim + khalf;
        v8f c0 = {0.f, 0.f, 0.f, 0.f, 0.f, 0.f, 0.f, 0.f};
#pragma unroll 8
        for (int k = 0; k < Kdim; k += 4) {
            v2f a = *(const v2f*)(alsA + k);
            if (HASA2) a += *(const v2f*)(alsA2 + k);
            v2f b0 = *(const v2f*)(w0 + k);
            c0 = __builtin_amdgcn_wmma_f32_16x16x4_f32(
                     false, a, false, b0, (short)0, c0, false, false);
        }
        epilogue(c0, tn);
    }
}

// ---------------------------------------------------------------------------
// Elementwise add: qk = tgt + query_pos
// ---------------------------------------------------------------------------
__global__ void add_kernel(const float* __restrict__ a, const float* __restrict__ b,
                           float* __restrict__ c, int n)
{
    int i = blockIdx.x * 256 + threadIdx.x;
    if (i < n) c[i] = a[i] + b[i];
}

// ---------------------------------------------------------------------------
// Self-attention, one block per (b,h). K/V staged in LDS (76.8KB of the
// 320KB WGP LDS). Online (flash) softmax, one query row per thread.
// qkv rows are l*16+b, cols [Q|K|V] of 256 each.
// ---------------------------------------------------------------------------
__global__ __launch_bounds__(256) void attn_kernel(const float* __restrict__ qkv,
                                                   float* __restrict__ out)
{
    extern __shared__ float lds[];
    float* Ks = lds;
    float* Vs = lds + 300 * 32;
    const int h = blockIdx.x & 7;
    const int b = blockIdx.x >> 3;
    const int tid = threadIdx.x;

    for (int i = tid; i < 300 * 32; i += 256) {
        int m = i >> 5, d = i & 31;
        long base = ((long)m * 16 + b) * 768 + h * 32 + d;
        Ks[i] = qkv[base + 256];
        Vs[i] = qkv[base + 512];
    }
    __syncthreads();

    const float scale = 0.17677669529663687f;   // 1/sqrt(32)
    for (int l = tid; l < 300; l += 256) {
        float qv[32], acc[32];
        long qbase = ((long)l * 16 + b) * 768 + h * 32;
#pragma unroll
        for (int d = 0; d < 32; d++) { qv[d] = qkv[qbase + d] * scale; acc[d] = 0.f; }
        float mx = -1e30f, sum = 0.f;
        for (int m = 0; m < 300; m++) {
            const float* kr = Ks + m * 32;
            float s = 0.f;
#pragma unroll
            for (int d = 0; d < 32; d++) s += qv[d] * kr[d];
            float mn = fmaxf(mx, s);
            float corr = __expf(mx - mn);
            float p = __expf(s - mn);
            sum = sum * corr + p;
            const float* vr = Vs + m * 32;
#pragma unroll
            for (int d = 0; d < 32; d++) acc[d] = acc[d] * corr + p * vr[d];
            mx = mn;
        }
        float inv = 1.f / sum;
        long obase = ((long)l * 16 + b) * 256 + h * 32;
#pragma unroll
        for (int d = 0; d < 32; d++) out[obase + d] = acc[d] * inv;
    }
}

// ---------------------------------------------------------------------------
// LayerNorm over C=256 with residual add, one row per wave32, shuffle reduce.
// mode 0: out_row = row;  mode 1: row = l*16+b -> out b*300+l (LxB -> BxL)
// mode 2: row = b*300+l -> out l*16+b (BxL -> LxB)
// ---------------------------------------------------------------------------
__global__ __launch_bounds__(256) void ln_kernel(
    const float* __restrict__ x1, const float* __restrict__ x2,
    const float* __restrict__ g, const float* __restrict__ bb,
    float* __restrict__ out, int rows, int mode)
{
    const int lane = threadIdx.x & 31;
    const int wave = threadIdx.x >> 5;
    const int row = blockIdx.x * 8 + wave;
    if (row >= rows) return;
    const float* p1 = x1 + (long)row * 256;
    const float* p2 = x2 + (long)row * 256;
    float v[8];
    float s = 0.f;
#pragma unroll
    for (int j = 0; j < 8; j++) { int c = lane + 32 * j; v[j] = p1[c] + p2[c]; s += v[j]; }
#pragma unroll
    for (int o = 16; o > 0; o >>= 1) s += __shfl_xor(s, o, 32);
    float mu = s * (1.0f / 256.0f);
    float q = 0.f;
#pragma unroll
    for (int j = 0; j < 8; j++) { float d = v[j] - mu; q += d * d; }
#pragma unroll
    for (int o = 16; o > 0; o >>= 1) q += __shfl_xor(q, o, 32);
    float rstd = rsqrtf(q * (1.0f / 256.0f) + 1e-5f);
    long orow;
    if (mode == 0) orow = row;
    else if (mode == 1) { int b = row & 15, l = row >> 4; orow = (long)b * 300 + l; }
    else { int b = row / 300, l = row % 300; orow = (long)l * 16 + b; }
    float* po = out + orow * 256;
#pragma unroll
    for (int j = 0; j < 8; j++) {
        int c = lane + 32 * j;
        po[c] = (v[j] - mu) * rstd * g[c] + bb[c];
    }
}

// ---------------------------------------------------------------------------
// Softmax over 16 (SCALES*K) per (b,q,head). One thread per group.
// ---------------------------------------------------------------------------
__global__ void asoftmax_kernel(const float* __restrict__ in, float* __restrict__ out,
                                int groups)
{
    int t = blockIdx.x * 256 + threadIdx.x;
    if (t >= groups) return;
    const float* p = in + (long)t * 16;
    float* o = out + (long)t * 16;
    float mx = p[0];
#pragma unroll
    for (int i = 1; i < 16; i++) mx = fmaxf(mx, p[i]);
    float e[16], s = 0.f;
#pragma unroll
    for (int i = 0; i < 16; i++) { e[i] = __expf(p[i] - mx); s += e[i]; }
    float inv = 1.f / s;
#pragma unroll
    for (int i = 0; i < 16; i++) o[i] = e[i] * inv;
}

// ---------------------------------------------------------------------------
// Deformable sampling: one wave per (b,q,h), lane = channel d (DH=32).
// Each corner gather is 32 contiguous floats -> one 128B coalesced request.
// ---------------------------------------------------------------------------
__global__ __launch_bounds__(256) void deform_sample_kernel(
    const float* __restrict__ off, const float* __restrict__ Aw,
    const float* __restrict__ vals, const float* __restrict__ ref_point,
    float* __restrict__ out)
{
    const int HWs[4] = {16, 32, 64, 128};
    const long rowOff[4] = {0, 4096, 20480, 86016};  // token offsets per level
    const int lane = threadIdx.x & 31;
    const int wave = threadIdx.x >> 5;
    const int unit = blockIdx.x * 8 + wave;           // 38400 = 16*300*8
    const int h = unit & 7;
    const int q = (unit >> 3) % 300;
    const int b = unit / 2400;
    const long row = (long)b * 300 + q;

    const float rx = ref_point[((long)q * 16 + b) * 2 + 0];
    const float ry = ref_point[((long)q * 16 + b) * 2 + 1];

    float acc = 0.f;
#pragma unroll
    for (int l = 0; l < 4; l++) {
        const int HW = HWs[l];
        const float* vb = vals + (rowOff[l] + (long)b * HW * HW) * 256 + h * 32 + lane;
#pragma unroll
        for (int k = 0; k < 4; k++) {
            const int ch = ((h * 4 + l) * 4 + k) * 2;
            float px = rx * (float)(HW - 1) + off[row * 256 + ch];
            float py = ry * (float)(HW - 1) + off[row * 256 + ch + 1];
            float a  = Aw[row * 128 + h * 16 + l * 4 + k];
            float x0 = floorf(px), y0 = floorf(py);
            float wx = px - x0, wy = py - y0;
            float sval = 0.f;
#pragma unroll
            for (int cy = 0; cy < 2; cy++)
#pragma unroll
                for (int cx = 0; cx < 2; cx++) {
                    float xf = x0 + (float)cx, yf = y0 + (float)cy;
                    float w = (cx ? wx : 1.f - wx) * (cy ? wy : 1.f - wy);
                    if (xf >= 0.f && xf <= (float)(HW - 1) &&
                        yf >= 0.f && yf <= (float)(HW - 1)) {
                        int xi = (int)xf, yi = (int)yf;
                        sval += w * vb[((long)yi * HW + xi) * 256];
                    }
                }
            acc += a * sval;
        }
    }
    out[row * 256 + h * 32 + lane] = acc;
}

// ---------------------------------------------------------------------------
extern "C" void kernel_launch(void* const* d_in, const int* in_sizes, int n_in,
                              void* d_out, int out_size, void* d_ws, size_t ws_size,
                              hipStream_t stream)
{
    const float* tgt       = (const float*)d_in[0];
    const float* query_pos = (const float*)d_in[1];
    const float* ref_point = (const float*)d_in[2];
    const float* mem[4] = {(const float*)d_in[3], (const float*)d_in[5],
                           (const float*)d_in[7], (const float*)d_in[9]};
    const float* pos[4] = {(const float*)d_in[4], (const float*)d_in[6],
                           (const float*)d_in[8], (const float*)d_in[10]};
    const float* Wqkv = (const float*)d_in[11]; const float* bqkv = (const float*)d_in[12];
    const float* Wo   = (const float*)d_in[13]; const float* bo   = (const float*)d_in[14];
    const float* W_off= (const float*)d_in[15]; const float* b_off= (const float*)d_in[16];
    const float* W_A  = (const float*)d_in[17]; const float* b_A  = (const float*)d_in[18];
    const float* W_v  = (const float*)d_in[19]; const float* b_v  = (const float*)d_in[20];
    const float* W_m  = (const float*)d_in[21]; const float* b_m  = (const float*)d_in[22];
    const float* W1   = (const float*)d_in[23]; const float* b1   = (const float*)d_in[24];
    const float* W2   = (const float*)d_in[25]; const float* b2   = (const float*)d_in[26];
    const float* ln1_g= (const float*)d_in[27]; const float* ln1_b= (const float*)d_in[28];
    const float* ln2_g= (const float*)d_in[29]; const float* ln2_b= (const float*)d_in[30];
    const float* ln3_g= (const float*)d_in[31]; const float* ln3_b= (const float*)d_in[32];

    const int ROWS = 4800;            // L*B = 300*16 token rows
    float* ws      = (float*)d_ws;
    float* qk      = ws;              // 4800*256
    float* qkv     = qk      + 1228800;   // 4800*768
    float* attn    = qkv     + 3686400;   // 4800*256
    float* proj    = attn    + 1228800;
    float* t1      = proj    + 1228800;   // [B,Lq,C]
    float* offb    = t1      + 1228800;
    float* Araw    = offb    + 1228800;   // 4800*128
    float* Asoft   = Araw    + 614400;
    float* sampled = Asoft   + 614400;
    float* t2m     = sampled + 1228800;
    float* tq2     = t2m     + 1228800;
    float* hbuf    = tq2     + 1228800;   // 4800*2048
    float* t2b     = hbuf    + 9830400;
    float* vals    = t2b     + 1228800;   // 348160*256

    const size_t lds256   = (size_t)16 * (256  + GPAD) * sizeof(float);
    const size_t lds256x2 = 2 * lds256;                       // A + A2 tiles
    const size_t lds2048  = (size_t)16 * (2048 + GPAD) * sizeof(float);

    // --- Stage A: self-attention block ---
    add_kernel<<<4800, 256, 0, stream>>>(tgt, query_pos, qk, 1228800);
    gemm_wmma_f32<false><<<ROWS / 16, 256, lds256, stream>>>(qk, nullptr, Wqkv, bqkv, qkv,
                                                             ROWS, 256, 768, 0);
    attn_kernel<<<128, 256, (size_t)2 * 300 * 32 * sizeof(float), stream>>>(qkv, attn);
    gemm_wmma_f32<false><<<ROWS / 16, 256, lds256, stream>>>(attn, nullptr, Wo, bo, proj,
                                                             ROWS, 256, 256, 0);
    ln_kernel<<<600, 256, 0, stream>>>(tgt, proj, ln1_g, ln1_b, t1, ROWS, 1);

    // --- Stage B: deformable attention ---
    const int HWs[4] = {16, 32, 64, 128};
    long voff = 0;
    for (int l = 0; l < 4; l++) {
        int Mr = 16 * HWs[l] * HWs[l];
        gemm_wmma_f32<true><<<Mr / 16, 256, lds256x2, stream>>>(mem[l], pos[l], W_v, b_v,
                                                                vals + voff * 256, Mr, 256, 256, 0);
        voff += Mr;
    }
    gemm_wmma_f32<false><<<ROWS / 16, 256, lds256, stream>>>(t1, nullptr, W_off, b_off, offb,
                                                             ROWS, 256, 256, 0);
    gemm_wmma_f32<false><<<ROWS / 16, 256, lds256, stream>>>(t1, nullptr, W_A, b_A, Araw,
                                                             ROWS, 256, 128, 0);
    asoftmax_kernel<<<150, 256, 0, stream>>>(Araw, Asoft, 38400);
    deform_sample_kernel<<<4800, 256, 0, stream>>>(offb, Asoft, vals, ref_point, sampled);
    gemm_wmma_f32<false><<<ROWS / 16, 256, lds256, stream>>>(sampled, nullptr, W_m, b_m, t2m,
                                                             ROWS, 256, 256, 0);
    ln_kernel<<<600, 256, 0, stream>>>(t1, t2m, ln2_g, ln2_b, tq2, ROWS, 0);

    // --- Stage C: FFN ---
    gemm_wmma_f32<false><<<ROWS / 16, 256, lds256, stream>>>(tq2, nullptr, W1, b1, hbuf,
                                                             ROWS, 256, 2048, 1);
    gemm_wmma_f32<false><<<ROWS / 16, 256, lds2048, stream>>>(hbuf, nullptr, W2, b2, t2b,
                                                              ROWS, 2048, 256, 0);
    ln_kernel<<<600, 256, 0, stream>>>(tq2, t2b, ln3_g, ln3_b, (float*)d_out, ROWS, 2);
}